// Quantizer_13408887898748
// MI455X (gfx1250) — compile-verified
//
#include <hip/hip_runtime.h>
#include <hip/hip_bf16.h>
#include <math.h>

typedef __attribute__((ext_vector_type(2))) float v2f;
typedef __attribute__((ext_vector_type(8))) float v8f;

#define DDIM   256
#define KCODES 1024
#define NVEC   32768
#define THW    4096
#define DECAY  0.99f
#define ALPHA  1e-5f
#define NUMEL  8388608.0f   // B*D*T*H*W

// d_out layout (floats): [0]=loss, [1..8388609)=quantized, [..+32768)=idx, [..+262144)=new_embed
#define OUT_QUANT 1
#define OUT_IDX   (1 + 8388608)
#define OUT_NE    (1 + 8388608 + 32768)

// workspace layout (floats)
#define WS_E2    0          // 1024
#define WS_NCS   1024       // 1024
#define WS_NSUM  2048       // 1
#define WS_CSB   4096       // 1024  (zeroed region starts here)
#define WS_LOSS  5120       // 1
#define WS_DW    8192       // 262144 (contiguous with zeroed region)
#define WS_IDX   270336     // 32768 ints
#define WS_ZERO_N (WS_IDX - WS_CSB)   // 266240 floats zeroed each call

// LDS tile layout: pair-major. float offset = (d>>1)*32 + 2*c + (d&1).
// -> each lane's WMMA-B operand (K=d0,d0+1 at column c) is one aligned 8-byte chunk;
//    a wave's 32 b64 reads cover 64 consecutive dwords = all 64 banks exactly once.
#define ES_BUF 4096   // 16 codes * 256 d floats per buffer

__global__ void zero_kernel(float* p, int n) {
    int i = blockIdx.x * 256 + threadIdx.x;
    if (i < n) p[i] = 0.0f;
}

// ||e_k||^2 : one block per code, coalesced read, LDS reduce
__global__ void e2_kernel(const float* __restrict__ embed, float* __restrict__ e2) {
    __shared__ float red[256];
    const int k = blockIdx.x, t = threadIdx.x;
    float v = embed[k * DDIM + t];
    red[t] = v * v;
    __syncthreads();
    for (int s = 128; s > 0; s >>= 1) {
        if (t < s) red[t] += red[t + s];
        __syncthreads();
    }
    if (t == 0) e2[k] = red[0];
}

// Async-stage one 16-code embed tile into an LDS buffer in pair-major layout.
// thread t: u = t&127 owns d-pair (2u,2u+1); half = t>>7 owns codes half*8..+7.
// Global side: consecutive lanes -> consecutive 8B chunks (coalesced b64s).
// LDS side contiguous per load. 8 x GLOBAL_LOAD_ASYNC_TO_LDS_B64 per thread.
__device__ __forceinline__ void stage_tile_async(const float* __restrict__ embed,
                                                 const float* ldsbuf, int ct, int t) {
    const int u = t & 127, half = t >> 7;
    const float* src = embed + ct * (16 * DDIM) + (half * 8) * DDIM + (u << 1);
    const float* dst = ldsbuf + u * 32 + half * 16;
#pragma unroll
    for (int i = 0; i < 8; ++i) {
        unsigned lds_off = (unsigned)(size_t)(const void*)(dst + 2 * i);
        unsigned long long ga = (unsigned long long)(size_t)(const void*)(src + i * DDIM);
        asm volatile("global_load_async_to_lds_b64 %0, %1, off"
                     :: "v"(lds_off), "v"(ga)
                     : "memory");
    }
}

#define WAIT_ASYNC() asm volatile("s_wait_asynccnt 0x0" ::: "memory")

// Nearest-code search via V_WMMA_F32_16X16X4_F32.
// Block = 256 thr = 8 waves; wave w owns rows [blk*128 + 16w, +16).
// A tile (16 rows x 256 d of x) lives in 128 VGPRs, loaded from HBM exactly once.
// B tiles (16 codes x 256 d of embed) stream through a double-buffered LDS pipeline
// fed by GLOBAL_LOAD_ASYNC_TO_LDS_B64 (ASYNCcnt), overlapping staging with the WMMAs.
__global__ void __launch_bounds__(256)
argmin_wmma_kernel(const float* __restrict__ inputs, const float* __restrict__ embed,
                   const float* __restrict__ e2, int* __restrict__ idxi,
                   float* __restrict__ out_idx) {
    __shared__ float es[2][ES_BUF];    // 32768 B

    const int lane = threadIdx.x & 31;
    const int w    = threadIdx.x >> 5;     // wave 0..7
    const int c    = lane & 15;            // code column in tile / row id for A
    const int hi   = lane >> 4;            // 0: K pair {0,1}, 1: K pair {2,3}
    const int koff = hi << 1;
    const int t    = threadIdx.x;          // staging id

    const int rowTile = blockIdx.x * 128 + w * 16;   // 4096 % 128 == 0 -> same b for block
    const int b = rowTile >> 12;
    const int s = (rowTile & (THW - 1)) + c;         // this lane's spatial index (A row M=c)
    const float* xbase = inputs + (size_t)b * (DDIM * THW) + s;

    // kick off tile 0 before touching A so the DMA overlaps the A-fragment fill
    stage_tile_async(embed, &es[0][0], 0, t);

    // ---- A fragments in registers: a[i] covers K = 4i+koff, 4i+koff+1 ----
    v2f a[64];
#pragma unroll
    for (int i = 0; i < 64; ++i) {
        const int d0 = 4 * i + koff;
        a[i].x = xbase[(size_t)d0 * THW];
        a[i].y = xbase[(size_t)(d0 + 1) * THW];
    }

    float bd[8];
    int   bk[8];
#pragma unroll
    for (int v = 0; v < 8; ++v) { bd[v] = INFINITY; bk[v] = 0; }

    const int bbase = hi * 16 + c;   // v2f units: lane's slot within a 32-v2f step group

    for (int ct = 0; ct < KCODES / 16; ++ct) {
        const int buf = ct & 1;
        WAIT_ASYNC();          // my share of tile ct has landed in LDS
        __syncthreads();       // everyone's share landed AND everyone finished tile ct-1
        if (ct + 1 < KCODES / 16) {
            stage_tile_async(embed, &es[buf ^ 1][0], ct + 1, t);   // overlaps with WMMAs below
            if (ct + 2 < KCODES / 16)                              // warm L2 one tile further out
                __builtin_prefetch(embed + (ct + 2) * (16 * DDIM) + t * 16, 0, 1);
        }

        const v2f*  ebp = (const v2f*)&es[buf][0];
        const float e2c = e2[ct * 16 + c];

        v8f acc0 = {}, acc1 = {};
#pragma unroll
        for (int i = 0; i < 64; i += 2) {
            const v2f b0 = ebp[bbase + 32 * i];          // (2i+hi)*16 + c : one ds_load_b64
            acc0 = __builtin_amdgcn_wmma_f32_16x16x4_f32(false, a[i], false, b0,
                                                         (short)0, acc0, false, false);
            const v2f b1 = ebp[bbase + 32 * (i + 1)];
            acc1 = __builtin_amdgcn_wmma_f32_16x16x4_f32(false, a[i + 1], false, b1,
                                                         (short)0, acc1, false, false);
        }

        const int kcode = ct * 16 + c;
#pragma unroll
        for (int v = 0; v < 8; ++v) {
            const float d2 = e2c - 2.0f * (acc0[v] + acc1[v]);   // row-constant ||x||^2 dropped
            if (d2 < bd[v]) { bd[v] = d2; bk[v] = kcode; }        // ct ascending -> first min kept
        }
    }

    // cross-lane argmin over the 16 code-columns (C layout: lanes 0-15 are M=v, 16-31 are M=v+8)
#pragma unroll
    for (int v = 0; v < 8; ++v) {
        float d = bd[v];
        int   k = bk[v];
        for (int m = 1; m < 16; m <<= 1) {
            const float od = __shfl_xor(d, m, 16);
            const int   ok = __shfl_xor(k, m, 16);
            if (od < d || (od == d && ok < k)) { d = od; k = ok; }
        }
        if (c == 0) {
            const int n = rowTile + hi * 8 + v;
            idxi[n]    = k;
            out_idx[n] = (float)k;
        }
    }
}

// quant = embed[idx] in [B,D,T,H,W] layout, commit-loss partial sum, and EMA scatters.
__global__ void quant_loss_scatter_kernel(const float* __restrict__ inputs,
                                          const float* __restrict__ embed,
                                          const int* __restrict__ idxi,
                                          float* __restrict__ out_quant,
                                          float* __restrict__ dw,
                                          float* __restrict__ cs_batch,
                                          float* __restrict__ loss_acc) {
    const int n  = blockIdx.x * 256 + threadIdx.x;
    const int id = idxi[n];
    const int b  = n >> 12;
    const int s  = n & (THW - 1);
    atomicAdd(&cs_batch[id], 1.0f);

    const float* erow = embed + (size_t)id * DDIM;
    const float* xrow = inputs + (size_t)b * (DDIM * THW) + s;
    float*       qrow = out_quant + (size_t)b * (DDIM * THW) + s;
    float lsum = 0.0f;
    for (int d = 0; d < DDIM; ++d) {
        const float q  = erow[d];
        const float xv = xrow[(size_t)d * THW];
        qrow[(size_t)d * THW] = q;                // quantized == quant (straight-through fwd)
        const float df = q - xv;
        lsum += df * df;
        atomicAdd(&dw[id * DDIM + d], xv);
    }

    __shared__ float red[256];
    red[threadIdx.x] = lsum;
    __syncthreads();
    for (int s2 = 128; s2 > 0; s2 >>= 1) {
        if (threadIdx.x < s2) red[threadIdx.x] += red[threadIdx.x + s2];
        __syncthreads();
    }
    if (threadIdx.x == 0) atomicAdd(loss_acc, red[0]);
}

// new_cluster_size + n = sum(new_cluster_size); also emits the scalar loss.
__global__ void ema_kernel(const float* __restrict__ cluster_size,
                           const float* __restrict__ cs_batch,
                           float* __restrict__ ncs, float* __restrict__ nsum,
                           const float* __restrict__ loss_acc, float* __restrict__ out_loss) {
    __shared__ float red[1024];
    const int k = threadIdx.x;
    const float v = cluster_size[k] * DECAY + (1.0f - DECAY) * cs_batch[k];
    ncs[k] = v;
    red[k] = v;
    __syncthreads();
    for (int s = 512; s > 0; s >>= 1) {
        if (k < s) red[k] += red[k + s];
        __syncthreads();
    }
    if (k == 0) {
        nsum[0]     = red[0];
        out_loss[0] = 0.25f * loss_acc[0] / NUMEL;
    }
}

__global__ void newembed_kernel(const float* __restrict__ embed_avg,
                                const float* __restrict__ dw,
                                const float* __restrict__ ncs,
                                const float* __restrict__ nsum,
                                float* __restrict__ out_ne) {
    const int j = blockIdx.x * 1024 + threadIdx.x;
    const int k = j >> 8;
    const float avg = embed_avg[j] * DECAY + (1.0f - DECAY) * dw[j];
    const float nn  = nsum[0];
    const float csk = nn * ((ncs[k] + ALPHA) / (nn + (float)KCODES * ALPHA));
    out_ne[j] = avg / csk;
}

extern "C" void kernel_launch(void* const* d_in, const int* in_sizes, int n_in,
                              void* d_out, int out_size, void* d_ws, size_t ws_size,
                              hipStream_t stream) {
    const float* inputs       = (const float*)d_in[0];
    const float* embed        = (const float*)d_in[1];
    const float* embed_avg    = (const float*)d_in[2];
    const float* cluster_size = (const float*)d_in[3];
    float* out = (float*)d_out;
    float* wsf = (float*)d_ws;

    float* e2       = wsf + WS_E2;
    float* ncs      = wsf + WS_NCS;
    float* nsum     = wsf + WS_NSUM;
    float* cs_batch = wsf + WS_CSB;
    float* loss_acc = wsf + WS_LOSS;
    float* dw       = wsf + WS_DW;
    int*   idxi     = (int*)(wsf + WS_IDX);

    zero_kernel<<<(WS_ZERO_N + 255) / 256, 256, 0, stream>>>(wsf + WS_CSB, WS_ZERO_N);
    e2_kernel<<<KCODES, 256, 0, stream>>>(embed, e2);
    argmin_wmma_kernel<<<NVEC / 128, 256, 0, stream>>>(inputs, embed, e2, idxi, out + OUT_IDX);
    quant_loss_scatter_kernel<<<NVEC / 256, 256, 0, stream>>>(inputs, embed, idxi,
                                                              out + OUT_QUANT, dw, cs_batch, loss_acc);
    ema_kernel<<<1, 1024, 0, stream>>>(cluster_size, cs_batch, ncs, nsum, loss_acc, out);
    newembed_kernel<<<KCODES * DDIM / 1024, 1024, 0, stream>>>(embed_avg, dw, ncs, nsum, out + OUT_NE);
}